// GATMultiHeadBlock_37297495999115
// MI455X (gfx1250) — compile-verified
//
#include <hip/hip_runtime.h>

typedef __attribute__((ext_vector_type(2))) float v2f;
typedef __attribute__((ext_vector_type(8))) float v8f;

#define NODES 100000
#define NEDGE 1600000
#define EPLUS (NEDGE + NODES)   // edges + self loops = 1,700,000
#define DFEAT 64
#define HEADS 4
#define HD (HEADS * DFEAT)      // 256
#define NEG_SLOPE 0.2f

// ---------------------------------------------------------------------------
// GEMM1: H[N,256] = X[N,64] @ W[64,256] using V_WMMA_F32_16X16X4_F32.
// One wave computes one 16x16 output tile; K-loop of 64/4 = 16 WMMAs.
// A 16x4 layout: lanes 0-15 rows M=0..15 holding {K0,K1}; lanes 16-31 {K2,K3}.
// B 4x16 layout (mirror): lanes 0-15 cols N=0..15 holding {K0,K1}; 16-31 {K2,K3}.
// D 16x16 layout: VGPR g -> row g (lanes 0-15) / row g+8 (lanes 16-31), col=lane&15.
// ---------------------------------------------------------------------------
__global__ __launch_bounds__(128)
void k_gemm_xw(const float* __restrict__ X, const float* __restrict__ W,
               float* __restrict__ H) {
  int wave = blockIdx.x * (blockDim.x >> 5) + (threadIdx.x >> 5);
  int lane = threadIdx.x & 31;
  const int ntiles = HD / 16;              // 16 column tiles
  int tm = wave / ntiles;
  int tn = wave % ntiles;
  int row0 = tm * 16, col0 = tn * 16;
  int r = lane & 15;
  int khalf = (lane >> 4) * 2;             // 0 or 2

  v8f acc = {};
  for (int k0 = 0; k0 < DFEAT; k0 += 4) {
    v2f a, b;
    const float* ap = X + (size_t)(row0 + r) * DFEAT + (k0 + khalf);
    a.x = ap[0];
    a.y = ap[1];
    const float* bp = W + (size_t)(k0 + khalf) * HD + (col0 + r);
    b.x = bp[0];
    b.y = bp[HD];
    acc = __builtin_amdgcn_wmma_f32_16x16x4_f32(false, a, false, b,
                                                (short)0, acc, false, false);
  }
  int drow = row0 + (lane >> 4) * 8;
  int dcol = col0 + (lane & 15);
#pragma unroll
  for (int g = 0; g < 8; ++g)
    H[(size_t)(drow + g) * HD + dcol] = acc[g];
}

// ---------------------------------------------------------------------------
// Per-(node,head) attention logit halves: a_src/a_dst = <h[n,h,:], att[h,:]>
// Row offset trick: (n*HEADS+hh)*DFEAT == n*HD + hh*DFEAT.
// ---------------------------------------------------------------------------
__global__ __launch_bounds__(256)
void k_att(const float* __restrict__ H, const float* __restrict__ att_src,
           const float* __restrict__ att_dst, float* __restrict__ a_src,
           float* __restrict__ a_dst) {
  int i = blockIdx.x * blockDim.x + threadIdx.x;     // i = n*HEADS + hh
  if (i >= NODES * HEADS) return;
  int hh = i & (HEADS - 1);
  const float4* hp = (const float4*)(H + (size_t)i * DFEAT);
  const float4* as = (const float4*)(att_src + hh * DFEAT);
  const float4* ad = (const float4*)(att_dst + hh * DFEAT);
  float ss = 0.f, sd = 0.f;
#pragma unroll 4
  for (int j = 0; j < DFEAT / 4; ++j) {
    float4 hv = hp[j], av = as[j], dv = ad[j];
    ss += hv.x * av.x + hv.y * av.y + hv.z * av.z + hv.w * av.w;
    sd += hv.x * dv.x + hv.y * dv.y + hv.z * dv.z + hv.w * dv.w;
  }
  a_src[i] = ss;
  a_dst[i] = sd;
}

__global__ __launch_bounds__(256)
void k_init(float* __restrict__ amax, float* __restrict__ denom,
            float* __restrict__ outagg) {
  int i = blockIdx.x * blockDim.x + threadIdx.x;
  if (i < NODES * HEADS) {
    amax[i] = -__builtin_inff();
    denom[i] = 0.f;
  }
  if (i < NODES * HD) outagg[i] = 0.f;
}

// Float atomic max via sign-split bit trick (init must be -inf = 0xFF800000).
__device__ __forceinline__ void atomicMaxF(float* addr, float v) {
  if (v >= 0.f)
    atomicMax((int*)addr, __float_as_int(v));
  else
    atomicMin((unsigned int*)addr, __float_as_uint(v));
}

// Pass 1 over edges: segment max of leaky-relu logits per (dst, head).
__global__ __launch_bounds__(256)
void k_edge_max(const int* __restrict__ ei, const float* __restrict__ a_src,
                const float* __restrict__ a_dst, float* __restrict__ amax) {
  long long i = (long long)blockIdx.x * blockDim.x + threadIdx.x;
  if (i >= (long long)EPLUS * HEADS) return;
  int hh = (int)(i & (HEADS - 1));
  long long e = i >> 2;
  int s, d;
  if (e < NEDGE) {
    s = ei[e];
    d = ei[NEDGE + e];
  } else {
    s = d = (int)(e - NEDGE);                       // self loop
  }
  float a = a_src[s * HEADS + hh] + a_dst[d * HEADS + hh];
  a = (a > 0.f) ? a : a * NEG_SLOPE;
  atomicMaxF(&amax[d * HEADS + hh], a);
}

// Pass 2 over edges (the hot loop): w = exp(alpha - amax[dst]);
// denom[dst,h] += w; outagg[dst,h,:] += w * h[src,h,:]  (unnormalized).
// 64 threads per edge: hh = (t%64)/16, 16 threads x float4 over D=64.
__global__ __launch_bounds__(256)
void k_edge_agg(const int* __restrict__ ei, const float* __restrict__ a_src,
                const float* __restrict__ a_dst, const float* __restrict__ amax,
                const float* __restrict__ H, float* __restrict__ denom,
                float* __restrict__ outagg) {
  long long t = (long long)blockIdx.x * blockDim.x + threadIdx.x;
  long long e = t >> 6;
  if (e >= EPLUS) return;
  int r = (int)(t & 63);
  int hh = r >> 4;
  int part = r & 15;
  int s, d;
  if (e < NEDGE) {
    s = ei[e];
    d = ei[NEDGE + e];
  } else {
    s = d = (int)(e - NEDGE);
  }
  float a = a_src[s * HEADS + hh] + a_dst[d * HEADS + hh];
  a = (a > 0.f) ? a : a * NEG_SLOPE;
  float w = __expf(a - amax[d * HEADS + hh]);
  if (part == 0) atomicAdd(&denom[d * HEADS + hh], w);
  const float4 hv =
      *(const float4*)(H + ((size_t)s * HEADS + hh) * DFEAT + part * 4);
  float* op = outagg + ((size_t)d * HEADS + hh) * DFEAT + part * 4;
  atomicAdd(op + 0, hv.x * w);
  atomicAdd(op + 1, hv.y * w);
  atomicAdd(op + 2, hv.z * w);
  atomicAdd(op + 3, hv.w * w);
}

// Normalize by softmax denom and add GAT output bias.
__global__ __launch_bounds__(256)
void k_norm(float* __restrict__ outagg, const float* __restrict__ denom,
            const float* __restrict__ bias) {
  int i = blockIdx.x * blockDim.x + threadIdx.x;
  if (i >= NODES * HD) return;
  int n = i >> 8;              // / HD
  int c = i & (HD - 1);
  int hh = c >> 6;             // / DFEAT
  outagg[i] = outagg[i] / (denom[n * HEADS + hh] + 1e-16f) + bias[c];
}

// ---------------------------------------------------------------------------
// GEMM2: OUT[N,64] = A[N,256] @ Wl[256,64] + bl, WMMA f32 16x16x4, K=256.
// ---------------------------------------------------------------------------
__global__ __launch_bounds__(128)
void k_gemm_out(const float* __restrict__ A, const float* __restrict__ B,
                const float* __restrict__ bl, float* __restrict__ C) {
  int wave = blockIdx.x * (blockDim.x >> 5) + (threadIdx.x >> 5);
  int lane = threadIdx.x & 31;
  const int ntiles = DFEAT / 16;           // 4 column tiles
  int tm = wave / ntiles;
  int tn = wave % ntiles;
  int row0 = tm * 16, col0 = tn * 16;
  int r = lane & 15;
  int khalf = (lane >> 4) * 2;

  v8f acc = {};
  for (int k0 = 0; k0 < HD; k0 += 4) {
    v2f a, b;
    const float* ap = A + (size_t)(row0 + r) * HD + (k0 + khalf);
    a.x = ap[0];
    a.y = ap[1];
    const float* bp = B + (size_t)(k0 + khalf) * DFEAT + (col0 + r);
    b.x = bp[0];
    b.y = bp[DFEAT];
    acc = __builtin_amdgcn_wmma_f32_16x16x4_f32(false, a, false, b,
                                                (short)0, acc, false, false);
  }
  int drow = row0 + (lane >> 4) * 8;
  int dcol = col0 + (lane & 15);
  float bb = bl[dcol];
#pragma unroll
  for (int g = 0; g < 8; ++g)
    C[(size_t)(drow + g) * DFEAT + dcol] = acc[g] + bb;
}

extern "C" void kernel_launch(void* const* d_in, const int* in_sizes, int n_in,
                              void* d_out, int out_size, void* d_ws,
                              size_t ws_size, hipStream_t stream) {
  const float* x       = (const float*)d_in[0];
  const int*   ei      = (const int*)d_in[1];   // [2,E] row-major: src then dst
  const float* W       = (const float*)d_in[2];
  const float* att_src = (const float*)d_in[3];
  const float* att_dst = (const float*)d_in[4];
  const float* bias    = (const float*)d_in[5];
  const float* Wl      = (const float*)d_in[6];
  const float* bl      = (const float*)d_in[7];
  float* out = (float*)d_out;

  // Workspace layout (floats): H | a_src | a_dst | amax | denom | outagg
  float* ws     = (float*)d_ws;
  float* Hbuf   = ws;                                   // N*HD
  float* a_src  = Hbuf + (size_t)NODES * HD;            // N*HEADS
  float* a_dst  = a_src + (size_t)NODES * HEADS;
  float* amax   = a_dst + (size_t)NODES * HEADS;
  float* denom  = amax + (size_t)NODES * HEADS;
  float* outagg = denom + (size_t)NODES * HEADS;        // N*HD

  // 1) H = X @ W  (6250*16 tiles, 4 waves/block)
  k_gemm_xw<<<(NODES / 16) * (HD / 16) / 4, 128, 0, stream>>>(x, W, Hbuf);

  // 2) per-(node,head) attention halves
  k_att<<<(NODES * HEADS + 255) / 256, 256, 0, stream>>>(Hbuf, att_src, att_dst,
                                                         a_src, a_dst);

  // 3) init amax/denom/outagg
  k_init<<<(NODES * HD + 255) / 256, 256, 0, stream>>>(amax, denom, outagg);

  // 4) segment max over incoming edges
  {
    long long nth = (long long)EPLUS * HEADS;
    int blocks = (int)((nth + 255) / 256);
    k_edge_max<<<blocks, 256, 0, stream>>>(ei, a_src, a_dst, amax);
  }

  // 5) fused exp + denom + weighted scatter-add (64 threads/edge)
  {
    long long nth = (long long)EPLUS * 64;
    int blocks = (int)((nth + 255) / 256);
    k_edge_agg<<<blocks, 256, 0, stream>>>(ei, a_src, a_dst, amax, Hbuf, denom,
                                           outagg);
  }

  // 6) normalize + GAT bias
  k_norm<<<(NODES * HD + 255) / 256, 256, 0, stream>>>(outagg, denom, bias);

  // 7) OUT = outagg @ Wl + bl  (6250*4 tiles, 4 waves/block)
  k_gemm_out<<<(NODES / 16) * (DFEAT / 16) / 4, 128, 0, stream>>>(outagg, Wl,
                                                                  bl, out);
}